// GraphVAE_14422500180364
// MI455X (gfx1250) — compile-verified
//
#include <hip/hip_runtime.h>
#include <hip/hip_bf16.h>
#include <math.h>

// GraphVAE forward on gfx1250.
// N=96, LATENT=256, OUT_DIM=4656, MPM_ITERS=50.
// Key identity: S[i,j,a,b] = P[i,j]*Q[a,b] (i!=j), S[i,i,a,a]=Sd[i,a], all >=0,
// so mpm step reduces to: m[j,a]=max_b Q[a,b]*x[j,b];  x = (x.*Sd + P@m)/||.||F
// The 96x96x96 f32 GEMM per iteration runs on V_WMMA_F32_16X16X4_F32.
// Q is symmetric (adj_rec is symmetric), so the max-mul reads Q[b*96+a]:
// consecutive lanes -> consecutive LDS addresses -> conflict-free.

typedef __attribute__((ext_vector_type(2))) float v2f;
typedef __attribute__((ext_vector_type(8))) float v8f;

#define NN 96
#define NSQ (NN * NN)           // 9216
#define LATENT 256
#define OUT_DIM 4656
#define MPM_ITERS 50
#define KCHUNK 128
#define ENC_BLOCKS (NSQ / KCHUNK)   // 72
#define DEC_BLOCKS 19               // ceil(4656/256)

// ---------------- Stage 1: encoder partial mat-vecs (memory-bound) -------------
__global__ void enc_partial(const float* __restrict__ h,
                            const float* __restrict__ We11,
                            const float* __restrict__ We12,
                            float* __restrict__ pmu, float* __restrict__ pls) {
    const int l  = threadIdx.x;             // 0..255 : output latent index
    const int kb = blockIdx.x * KCHUNK;
    float amu = 0.f, als = 0.f;
    for (int kk = 0; kk < KCHUNK; ++kk) {
        const int k = kb + kk;
        const float hv = h[k];                       // broadcast
        amu = fmaf(hv, We11[k * LATENT + l], amu);   // coalesced row read
        als = fmaf(hv, We12[k * LATENT + l], als);
    }
    pmu[blockIdx.x * LATENT + l] = amu;
    pls[blockIdx.x * LATENT + l] = als;
}

// ---------------- Stage 2: finish encoder, z, KL, hidden = relu(z@Wd1+b) -------
__global__ void enc_finish(const float* __restrict__ pmu, const float* __restrict__ pls,
                           const float* __restrict__ be11, const float* __restrict__ be12,
                           const float* __restrict__ eps,
                           const float* __restrict__ Wd1, const float* __restrict__ bd1,
                           float* __restrict__ klout, float* __restrict__ hid) {
    __shared__ float zsh[LATENT];
    __shared__ float red[LATENT];
    const int l = threadIdx.x;
    float mu = be11[l], ls = be12[l];
    for (int b = 0; b < ENC_BLOCKS; ++b) {
        mu += pmu[b * LATENT + l];
        ls += pls[b * LATENT + l];
    }
    zsh[l] = eps[l] * expf(0.5f * ls) + mu;
    red[l] = 1.0f + ls - mu * mu - expf(ls);
    __syncthreads();
    for (int s = LATENT / 2; s > 0; s >>= 1) {
        if (l < s) red[l] += red[l + s];
        __syncthreads();
    }
    if (l == 0) klout[0] = -0.5f * red[0] / (float)NSQ;
    float acc = bd1[l];
    for (int k = 0; k < LATENT; ++k)
        acc = fmaf(zsh[k], Wd1[k * LATENT + l], acc);
    hid[l] = fmaxf(acc, 0.0f);
}

// ---------------- Stage 3: y = hid@Wd2+b, sigmoid, BCE partials ----------------
__global__ void dec_out(const float* __restrict__ hid,
                        const float* __restrict__ Wd2, const float* __restrict__ bd2,
                        const float* __restrict__ adj,
                        float* __restrict__ outv, float* __restrict__ bce_part) {
    __shared__ float hsh[LATENT];
    __shared__ float red[256];
    const int tid = threadIdx.x;
    hsh[tid] = hid[tid];
    __syncthreads();
    const int o = blockIdx.x * 256 + tid;
    float term = 0.0f;
    if (o < OUT_DIM) {
        float y = bd2[o];
        for (int k = 0; k < LATENT; ++k)
            y = fmaf(hsh[k], Wd2[k * OUT_DIM + o], y);   // coalesced over o
        const float p = 1.0f / (1.0f + expf(-y));
        outv[o] = p;
        // linear triu index -> (i,j), i<=j
        int rem = o, i = 0;
        while (rem >= NN - i) { rem -= NN - i; ++i; }
        const int j = i + rem;
        const float a = adj[i * NN + j];
        // reference: _bce(adj_vec, out) -> input=adj, target=out
        const float li = fmaxf(logf(a), -100.0f);
        const float l1 = fmaxf(log1pf(-a), -100.0f);
        term = p * li + (1.0f - p) * l1;
    }
    red[tid] = term;
    __syncthreads();
    for (int s = 128; s > 0; s >>= 1) {
        if (tid < s) red[tid] += red[tid + s];
        __syncthreads();
    }
    if (tid == 0) bce_part[blockIdx.x] = red[0];
}

// ---------------- Stage 4: adj_rec, P, Q, Sd, final scalar loss ----------------
__global__ void build_mats(const float* __restrict__ outv, const float* __restrict__ adj,
                           const float* __restrict__ klv, const float* __restrict__ bce_part,
                           float* __restrict__ ar, float* __restrict__ P,
                           float* __restrict__ Q, float* __restrict__ Sd,
                           float* __restrict__ loss_out) {
    __shared__ float d[NN], dr[NN], ft[NN], ftr[NN];
    const int tid = threadIdx.x;
    for (int e = tid; e < NSQ; e += 256) {
        const int i = e / NN, j = e % NN;
        const int lo = i < j ? i : j;
        const int hi = i < j ? j : i;
        const int idx = lo * NN - lo * (lo - 1) / 2 + (hi - lo);
        ar[e] = outv[idx];                      // up + up.T - diag(diag(up))
    }
    __syncthreads();
    if (tid < NN) {
        float s0 = 0.f, s1 = 0.f;
        for (int j = 0; j < NN; ++j) {
            s0 += adj[tid * NN + j];
            s1 += ar[tid * NN + j];
        }
        ft[tid] = s0; ftr[tid] = s1;
        d[tid]  = adj[tid * (NN + 1)];
        dr[tid] = ar[tid * (NN + 1)];
    }
    __syncthreads();
    for (int e = tid; e < NSQ; e += 256) {
        const int i = e / NN, j = e % NN;
        P[e]  = (i == j) ? 0.0f : adj[e] * d[i] * d[j];
        Q[e]  = (i == j) ? 0.0f : ar[e] * dr[i] * dr[j];   // symmetric
        Sd[e] = d[i] * dr[j] / (fabsf(ft[i] - ftr[j]) + 1.0f);
    }
    if (tid == 0) {
        float r = 0.0f;
        for (int b = 0; b < DEC_BLOCKS; ++b) r += bce_part[b];
        loss_out[0] = -(r / (float)OUT_DIM) + klv[0];
    }
}

// ---------------- Stage 5: 50 MPM iterations, single WG, WMMA GEMM -------------
__global__ __launch_bounds__(1024)
void mpm_kernel(const float* __restrict__ Pg, const float* __restrict__ Qg,
                const float* __restrict__ Sdg, float* __restrict__ assign_out) {
    __shared__ float xbuf[NSQ];     // 36 KB
    __shared__ float mbuf[NSQ];     // 36 KB
    __shared__ float qbuf[NSQ];     // 36 KB (symmetric Q)
    __shared__ float red[1024];     //  4 KB   -> 112.6 KB total (< 320 KB WGP)
    const int tid  = threadIdx.x;
    const int lane = tid & 31;
    const int wave = tid >> 5;      // 32 waves
    const int hi   = lane >> 4;     // half-lane select (0/1)
    const int lj   = lane & 15;

    for (int e = tid; e < NSQ; e += 1024) {
        xbuf[e] = 1.0f / 96.0f;
        qbuf[e] = Qg[e];            // coalesced load, resident all 50 iters
    }
    __syncthreads();

    for (int it = 0; it < MPM_ITERS; ++it) {
        // m[j][a] = max_b Q[a][b]*x[j][b] = max_b Q[b][a]*x[j][b]  (Q symmetric)
        // lanes: consecutive a, same j -> qbuf read is lane-consecutive
        // (conflict-free banks), xbuf read is a broadcast.
        for (int e = tid; e < NSQ; e += 1024) {
            const int j = e / NN, a = e % NN;
            const float* xrow = xbuf + j * NN;
            float mx = 0.0f;
            for (int b = 0; b < NN; ++b)
                mx = fmaxf(mx, qbuf[b * NN + a] * xrow[b]);
            mbuf[e] = mx;
        }
        __syncthreads();

        // x_new = x .* Sd + P @ m   via 36 tiles of V_WMMA_F32_16X16X4_F32
        float ssq = 0.0f;
        for (int t = wave; t < 36; t += 32) {     // uniform per-wave: EXEC all ones
            const int I = t / 6, J = t % 6;
            v8f acc = {0.f, 0.f, 0.f, 0.f, 0.f, 0.f, 0.f, 0.f};
            const int arow = I * 16 + lj;         // A: M = lane&15
            const int bcol = J * 16 + lj;         // B: N = lane&15
            for (int kt = 0; kt < 24; ++kt) {
                const int k0 = kt * 4 + 2 * hi;   // K = v + 2*(lane>=16)
                v2f a = *(const v2f*)(Pg + arow * NN + k0);   // 8B-aligned
                v2f b;
                b.x = mbuf[(k0 + 0) * NN + bcol];
                b.y = mbuf[(k0 + 1) * NN + bcol];
#ifdef __HIP_DEVICE_COMPILE__
                acc = __builtin_amdgcn_wmma_f32_16x16x4_f32(
                    false, a, false, b, (short)0, acc, false, false);
#endif
            }
            for (int v = 0; v < 8; ++v) {         // C/D: M = v + 8*(lane>=16)
                const int row = I * 16 + v + 8 * hi;
                const int e = row * NN + bcol;
                const float xn = fmaf(xbuf[e], Sdg[e], acc[v]);
                xbuf[e] = xn;                     // in-place: only this thread touches e
                ssq = fmaf(xn, xn, ssq);
            }
        }
        red[tid] = ssq;
        __syncthreads();
        for (int s = 512; s > 0; s >>= 1) {       // deterministic tree
            if (tid < s) red[tid] += red[tid + s];
            __syncthreads();
        }
        const float inv = 1.0f / sqrtf(red[0]);
        for (int e = tid; e < NSQ; e += 1024) xbuf[e] *= inv;
        __syncthreads();
    }
    for (int e = tid; e < NSQ; e += 1024) assign_out[e] = xbuf[e];
}

// ---------------- host-side orchestration --------------------------------------
extern "C" void kernel_launch(void* const* d_in, const int* in_sizes, int n_in,
                              void* d_out, int out_size, void* d_ws, size_t ws_size,
                              hipStream_t stream) {
    (void)in_sizes; (void)n_in; (void)out_size; (void)ws_size;
    const float* in_feat = (const float*)d_in[0];   // (1,96,96)
    const float* adj     = (const float*)d_in[1];   // (1,96,96)
    const float* eps     = (const float*)d_in[2];   // (1,256)
    const float* We11    = (const float*)d_in[3];
    const float* be11    = (const float*)d_in[4];
    const float* We12    = (const float*)d_in[5];
    const float* be12    = (const float*)d_in[6];
    const float* Wd1     = (const float*)d_in[7];
    const float* bd1     = (const float*)d_in[8];
    const float* Wd2     = (const float*)d_in[9];
    const float* bd2     = (const float*)d_in[10];
    float* out = (float*)d_out;                     // [loss, assignment(9216)]
    float* ws  = (float*)d_ws;

    // workspace layout (float offsets, all even -> 8B alignment for v2f loads)
    float* pmu  = ws + 0;        // 72*256 = 18432
    float* pls  = ws + 18432;    // 18432
    float* klv  = ws + 36864;    // 2
    float* hid  = ws + 36866;    // 256
    float* outv = ws + 37122;    // 4656
    float* bce  = ws + 41778;    // 20
    float* ar   = ws + 41798;    // 9216
    float* P    = ws + 51014;    // 9216
    float* Q    = ws + 60230;    // 9216
    float* Sd   = ws + 69446;    // 9216  -> total 78662 floats (~315 KB)

    enc_partial<<<ENC_BLOCKS, 256, 0, stream>>>(in_feat, We11, We12, pmu, pls);
    enc_finish <<<1,          256, 0, stream>>>(pmu, pls, be11, be12, eps, Wd1, bd1, klv, hid);
    dec_out    <<<DEC_BLOCKS, 256, 0, stream>>>(hid, Wd2, bd2, adj, outv, bce);
    build_mats <<<1,          256, 0, stream>>>(outv, adj, klv, bce, ar, P, Q, Sd, out);
    mpm_kernel <<<1,         1024, 0, stream>>>(P, Q, Sd, out + 1);
}